// WideModel_87522843560495
// MI455X (gfx1250) — compile-verified
//
#include <hip/hip_runtime.h>
#include <cstdint>

// ---------------------------------------------------------------------------
// Wide (hashed multi-hot) linear model for MI455X / gfx1250.
//   out[b] = bias + sum_f sum_{distinct bins in row} W_f[bin]
// One wave32 per row. Row ids staged via CDNA5 async global->LDS DMA.
// Dedup via v_readlane broadcast loop (O(L) wave instructions).
// ---------------------------------------------------------------------------

#define NFEAT 8
#define LROW 50
#define LDS_STRIDE 64           // ints per feature slot (padded 50 -> 64)
#define ROWS_PER_BLOCK 8        // 8 waves * 32 lanes = 256 threads

struct WideArgs {
    const int*   x[NFEAT];      // [B, 50] int32, -1 = padding
    const float* w[NFEAT];      // weight tables
    const float* bias;          // [1]
    float*       out;           // [B]
    int          B;
};

// CDNA5 async copy: copies 4 bytes from the global address gptr to the LDS address lds_addr  (tracked by ASYNCcnt)
__device__ __forceinline__ void async_load_b32(unsigned lds_addr, const void* gptr) {
    asm volatile("global_load_async_to_lds_b32 %0, %1, off"
                 :: "v"(lds_addr),
                    "v"((unsigned long long)(uintptr_t)gptr)
                 : "memory");
}

__device__ __forceinline__ void wait_async_zero() {
    asm volatile("s_wait_asynccnt 0x0" ::: "memory");
}

// Contribution of one feature for this wave's row.
// Lane p owns positions p and p+32 (positions 50..63 of the slot are unused).
template <int BUCKET>
__device__ __forceinline__ float feature_contrib(const int* __restrict__ bp,
                                                 const float* __restrict__ W,
                                                 int lane) {
    int x0 = bp[lane];                                  // position lane
    int x1 = (lane < LROW - 32) ? bp[32 + lane] : -1;   // position 32+lane
    // Hash; sentinel BUCKET for padding (never equals a valid bin).
    int b0 = (x0 >= 0) ? (int)((unsigned)x0 % (unsigned)BUCKET) : BUCKET;
    int b1 = (x1 >= 0) ? (int)((unsigned)x1 % (unsigned)BUCKET) : BUCKET;

    int dup0 = 0, dup1 = 0;
    // Compare against positions 0..31 (all precede position 32+lane).
#pragma unroll
    for (int j = 0; j < 32; ++j) {
        int bj = __builtin_amdgcn_readlane(b0, j);      // wave-uniform broadcast
        dup0 |= ((j < lane) & (bj == b0));              // only earlier positions
        dup1 |= (bj == b1);
    }
    // Compare position 32+lane against positions 32..49.
#pragma unroll
    for (int j = 0; j < LROW - 32; ++j) {
        int bj = __builtin_amdgcn_readlane(b1, j);
        dup1 |= ((j < lane) & (bj == b1));
    }

    float r = 0.0f;
    if ((b0 < BUCKET) && !dup0) r += W[b0];             // predicated L2-hit gather
    if ((b1 < BUCKET) && !dup1) r += W[b1];
    return r;
}

__global__ __launch_bounds__(32 * ROWS_PER_BLOCK)
void wide_model_kernel(WideArgs a) {
    __shared__ int smem[ROWS_PER_BLOCK * NFEAT * LDS_STRIDE];   // 16 KB

    const int lane = threadIdx.x & 31;
    const int wid  = threadIdx.x >> 5;
    const int row  = blockIdx.x * ROWS_PER_BLOCK + wid;
    if (row >= a.B) return;                 // wave-uniform guard

    int* wbuf = &smem[wid * (NFEAT * LDS_STRIDE)];
    const int rowOff = row * LROW;

    // ---- Stage all 8 feature rows (400 ids) into LDS via async DMA ----
#pragma unroll
    for (int f = 0; f < NFEAT; ++f) {
        const int* gp = a.x[f] + rowOff;
        unsigned l0 = (unsigned)(uintptr_t)&wbuf[f * LDS_STRIDE + lane];
        async_load_b32(l0, gp + lane);                       // positions 0..31
        if (lane < LROW - 32)
            async_load_b32(l0 + 32u * 4u, gp + 32 + lane);   // positions 32..49
    }
    wait_async_zero();                      // ASYNCcnt == 0: LDS data visible

    // ---- Per-feature hashed-dedup-gather-sum ----
    float acc = 0.0f;
#pragma unroll 1
    for (int f = 0; f < 6; ++f)             // f0..f5: bucket 100000
        acc += feature_contrib<100000>(&wbuf[f * LDS_STRIDE], a.w[f], lane);
    acc += feature_contrib<1000000>(&wbuf[6 * LDS_STRIDE], a.w[6], lane);
    acc += feature_contrib<1000000>(&wbuf[7 * LDS_STRIDE], a.w[7], lane);

    // ---- wave32 butterfly reduction ----
#pragma unroll
    for (int k = 16; k >= 1; k >>= 1)
        acc += __shfl_xor(acc, k, 32);

    if (lane == 0)
        a.out[row] = acc + a.bias[0];
}

extern "C" void kernel_launch(void* const* d_in, const int* in_sizes, int n_in,
                              void* d_out, int out_size, void* d_ws, size_t ws_size,
                              hipStream_t stream) {
    (void)in_sizes; (void)n_in; (void)d_ws; (void)ws_size;

    WideArgs a;
    for (int i = 0; i < NFEAT; ++i) a.x[i] = (const int*)d_in[i];          // f0..f5,s0,s1
    for (int i = 0; i < NFEAT; ++i) a.w[i] = (const float*)d_in[NFEAT + i]; // W_*
    a.bias = (const float*)d_in[2 * NFEAT];
    a.out  = (float*)d_out;
    a.B    = out_size;                      // output is [B,1] fp32

    const int blocks = (a.B + ROWS_PER_BLOCK - 1) / ROWS_PER_BLOCK;
    wide_model_kernel<<<dim3(blocks), dim3(32 * ROWS_PER_BLOCK), 0, stream>>>(a);
}